// GATv2Conv_21741124452819
// MI455X (gfx1250) — compile-verified
//
#include <hip/hip_runtime.h>

// ---------------------------------------------------------------------------
// GATv2Conv on MI455X (gfx1250, wave32).
//   feat_src = x @ W_src + b_src   (WMMA f32 16x16x4 matrix core, full fp32)
//   feat_dst = x @ W_dst + b_dst
//   per-edge score -> segmented softmax over dst -> weighted scatter-add -> relu
// Roofline: GEMMs are ~6.5 GFLOP / ~150 MB -> negligible at 23.3 TB/s.
// Dominant cost is ~2.6 GB of per-edge feature gathers + ~218M f32 atomics;
// feat arrays (2 x 51.2 MB) fit in the 192 MB L2, so that traffic is
// L2-resident. Memory-system bound -> fp32 WMMA (no precision loss) is the
// right matrix-core choice.
// ---------------------------------------------------------------------------

typedef __attribute__((ext_vector_type(2))) float v2f;
typedef __attribute__((ext_vector_type(8))) float v8f;

#define F_IN 128
#define NH   4
#define ND   32
#define NEG_SLOPE 0.2f

// monotonic float<->uint encoding so integer atomicMax == float max
__device__ __forceinline__ unsigned enc_f32(float f) {
  unsigned u = __float_as_uint(f);
  return (u & 0x80000000u) ? ~u : (u | 0x80000000u);
}
__device__ __forceinline__ float dec_f32(unsigned u) {
  unsigned b = (u & 0x80000000u) ? (u & 0x7fffffffu) : ~u;
  return __uint_as_float(b);
}

// ---------------------------------------------------------------------------
// 0) init: zero output accumulator, denominators; smax := encoded(-inf) = 0
// ---------------------------------------------------------------------------
__global__ void __launch_bounds__(256)
gat_init(float* __restrict__ out, unsigned* __restrict__ smax,
         float* __restrict__ denom, int N) {
  int gid = blockIdx.x * blockDim.x + threadIdx.x;
  if (gid < N * F_IN) out[gid] = 0.0f;
  if (gid < N * NH) { smax[gid] = 0u; denom[gid] = 0.0f; }
}

// ---------------------------------------------------------------------------
// 1) GEMM via V_WMMA_F32_16X16X4_F32.
//    Block = 256 threads = 8 waves; each wave owns one 16-col tile of the
//    128-wide output; block covers one 16-row tile. blockIdx.y selects the
//    src / dst weight matrix.
//    A (16x4 f32): lane<16 holds row M=lane, K=k..k+1; lane>=16 holds K=k+2..3
//    B (4x16 f32): vgpr0 = rows K (lanes0-15) / K+2 (lanes16-31), col = lane%16
//    C/D (16x16 f32): vgpr i -> row i (lanes0-15) / i+8 (lanes16-31), col=lane%16
// ---------------------------------------------------------------------------
__global__ void __launch_bounds__(256)
gat_gemm_wmma(const float* __restrict__ x,
              const float* __restrict__ Wsrc, const float* __restrict__ bsrc,
              const float* __restrict__ Wdst, const float* __restrict__ bdst,
              float* __restrict__ fsrc, float* __restrict__ fdst, int N) {
  const float* W    = blockIdx.y ? Wdst : Wsrc;
  const float* bias = blockIdx.y ? bdst : bsrc;
  float*       outp = blockIdx.y ? fdst : fsrc;

  const int row0 = blockIdx.x * 16;
  const int wave = threadIdx.x >> 5;   // 0..7 -> 16-col tile
  const int lane = threadIdx.x & 31;
  const int half = lane >> 4;          // 0: K lo pair, 1: K hi pair
  const int mrow = lane & 15;
  const int n0   = wave * 16;

  // clamp the A-row read for a partial last tile (stores handled below)
  int arow = row0 + mrow; if (arow >= N) arow = N - 1;
  const float* xrow = x + (size_t)arow * F_IN;

  v8f c = {0.f, 0.f, 0.f, 0.f, 0.f, 0.f, 0.f, 0.f};
  #pragma unroll 8
  for (int k = 0; k < F_IN; k += 4) {
    const int ka = k + 2 * half;
    v2f a, b;
    a.x = xrow[ka];
    a.y = xrow[ka + 1];
    b.x = W[(size_t)ka * F_IN + n0 + mrow];
    b.y = W[(size_t)(ka + 1) * F_IN + n0 + mrow];
    c = __builtin_amdgcn_wmma_f32_16x16x4_f32(
        /*neg_a=*/false, a, /*neg_b=*/false, b,
        /*c_mod=*/(short)0, c, /*reuse_a=*/false, /*reuse_b=*/false);
  }

  const float bv = bias[n0 + mrow];
  float* po = outp + (size_t)(row0 + 8 * half) * F_IN + n0 + mrow;
  if (row0 + 16 <= N) {
    // full tile: branch-free store epilogue (the common case; N%16==0 here)
    #pragma unroll
    for (int i = 0; i < 8; ++i) po[(size_t)i * F_IN] = c[i] + bv;
  } else {
    #pragma unroll
    for (int i = 0; i < 8; ++i) {
      if (row0 + i + 8 * half < N) po[(size_t)i * F_IN] = c[i] + bv;
    }
  }
}

// ---------------------------------------------------------------------------
// 2) per-edge GATv2 score + segmented max (one wave32 per edge).
//    score[h] = sum_d attn[h,d] * leakyrelu(fsrc[s,h,d] + fdst[t,h,d])
// ---------------------------------------------------------------------------
__global__ void __launch_bounds__(256)
gat_score(const float* __restrict__ fsrc, const float* __restrict__ fdst,
          const float* __restrict__ attn,
          const int* __restrict__ src, const int* __restrict__ dst,
          int E, int N,
          float* __restrict__ sc, unsigned* __restrict__ smax) {
  const int wid  = (blockIdx.x * blockDim.x + threadIdx.x) >> 5;  // edge id
  const int lane = threadIdx.x & 31;
  const int EP = E + N;
  if (wid >= EP) return;

  int s, d;
  if (wid < E) { s = src[wid]; d = dst[wid]; } else { s = d = wid - E; }

  const float* ps = fsrc + (size_t)s * F_IN;
  const float* pd = fdst + (size_t)d * F_IN;

  float sum[NH];
  #pragma unroll
  for (int h = 0; h < NH; ++h) {
    const int idx = h * ND + lane;
    float v = ps[idx] + pd[idx];
    v = v > 0.0f ? v : NEG_SLOPE * v;
    sum[h] = v * attn[idx];
  }
  #pragma unroll
  for (int off = 16; off > 0; off >>= 1) {
    #pragma unroll
    for (int h = 0; h < NH; ++h) sum[h] += __shfl_xor(sum[h], off, 32);
  }
  if (lane == 0) {
    #pragma unroll
    for (int h = 0; h < NH; ++h) {
      sc[(size_t)wid * NH + h] = sum[h];
      atomicMax(&smax[(size_t)d * NH + h], enc_f32(sum[h]));
    }
  }
}

// ---------------------------------------------------------------------------
// 3) exp(score - max) + segmented sum (one thread per edge*head)
// ---------------------------------------------------------------------------
__global__ void __launch_bounds__(256)
gat_exp(const int* __restrict__ dst, int E, int N,
        float* __restrict__ sc, const unsigned* __restrict__ smax,
        float* __restrict__ denom) {
  const int gid = blockIdx.x * blockDim.x + threadIdx.x;
  const int EP = E + N;
  if (gid >= EP * NH) return;
  const int e = gid >> 2;         // NH == 4
  const int h = gid & 3;
  const int d = (e < E) ? dst[e] : (e - E);
  const float m  = dec_f32(smax[(size_t)d * NH + h]);
  const float es = expf(sc[gid] - m);
  sc[gid] = es;
  atomicAdd(&denom[(size_t)d * NH + h], es);
}

// ---------------------------------------------------------------------------
// 4) weighted scatter-aggregate (one wave32 per edge, 4 atomics per lane)
// ---------------------------------------------------------------------------
__global__ void __launch_bounds__(256)
gat_aggregate(const float* __restrict__ fsrc,
              const int* __restrict__ src, const int* __restrict__ dst,
              int E, int N,
              const float* __restrict__ sc, const float* __restrict__ denom,
              float* __restrict__ out) {
  const int wid  = (blockIdx.x * blockDim.x + threadIdx.x) >> 5;
  const int lane = threadIdx.x & 31;
  const int EP = E + N;
  if (wid >= EP) return;

  int s, d;
  if (wid < E) { s = src[wid]; d = dst[wid]; } else { s = d = wid - E; }

  const float* ps = fsrc + (size_t)s * F_IN;
  float*       po = out + (size_t)d * F_IN;
  #pragma unroll
  for (int h = 0; h < NH; ++h) {
    const float alpha = sc[(size_t)wid * NH + h] / denom[(size_t)d * NH + h];
    const int idx = h * ND + lane;
    atomicAdd(&po[idx], alpha * ps[idx]);
  }
}

// ---------------------------------------------------------------------------
// 5) in-place ReLU on the output
// ---------------------------------------------------------------------------
__global__ void __launch_bounds__(256)
gat_relu(float* __restrict__ out, int n) {
  const int gid = blockIdx.x * blockDim.x + threadIdx.x;
  if (gid < n) out[gid] = fmaxf(out[gid], 0.0f);
}

// ---------------------------------------------------------------------------
extern "C" void kernel_launch(void* const* d_in, const int* in_sizes, int n_in,
                              void* d_out, int out_size, void* d_ws, size_t ws_size,
                              hipStream_t stream) {
  (void)n_in; (void)out_size; (void)ws_size;
  const float* x    = (const float*)d_in[0];
  const int*   src  = (const int*)  d_in[1];
  const int*   dst  = (const int*)  d_in[2];
  const float* Wsrc = (const float*)d_in[3];
  const float* bsrc = (const float*)d_in[4];
  const float* Wdst = (const float*)d_in[5];
  const float* bdst = (const float*)d_in[6];
  const float* attn = (const float*)d_in[7];

  const int N  = in_sizes[0] / F_IN;   // 100000
  const int E  = in_sizes[1];          // 1600000
  const int EP = E + N;                // edges + self loops
  float* out = (float*)d_out;

  // workspace partition (~133 MB total)
  char* ws = (char*)d_ws;
  float*    fsrc  = (float*)ws;    ws += (size_t)N * F_IN * sizeof(float);
  float*    fdst  = (float*)ws;    ws += (size_t)N * F_IN * sizeof(float);
  float*    sc    = (float*)ws;    ws += (size_t)EP * NH * sizeof(float);
  unsigned* smax  = (unsigned*)ws; ws += (size_t)N * NH * sizeof(unsigned);
  float*    denom = (float*)ws;

  const int TB = 256;

  gat_init<<<(N * F_IN + TB - 1) / TB, TB, 0, stream>>>(out, smax, denom, N);

  dim3 ggrid((N + 15) / 16, 2, 1);
  gat_gemm_wmma<<<ggrid, TB, 0, stream>>>(x, Wsrc, bsrc, Wdst, bdst, fsrc, fdst, N);

  gat_score<<<(EP + 7) / 8, TB, 0, stream>>>(fsrc, fdst, attn, src, dst, E, N, sc, smax);

  gat_exp<<<(EP * NH + TB - 1) / TB, TB, 0, stream>>>(dst, E, N, sc, smax, denom);

  gat_aggregate<<<(EP + 7) / 8, TB, 0, stream>>>(fsrc, src, dst, E, N, sc, denom, out);

  gat_relu<<<(N * F_IN + TB - 1) / TB, TB, 0, stream>>>(out, N * F_IN);
}